// QuantileQuantizationLayer_71528385347919
// MI455X (gfx1250) — compile-verified
//
#include <hip/hip_runtime.h>
#include <stdint.h>

// Problem constants (from reference): B=32768, F=512, T=255
#define NB 32768
#define NF 512
#define NT 255
#define FT 64               // features per block tile
#define THR_STRIDE 255      // odd stride -> conflict-free LDS banks across lanes
#define MID_STRIDE 257      // odd stride, holds 256 midpoints
#define BLOCK 256
#define ROWS_PER_BLOCK 1024 // grid.y = 32

// Generic->LDS byte offset: for shared pointers the low 32 bits of the flat
// address ARE the LDS offset (ISA 10.2).
__device__ __forceinline__ unsigned lds_off(const void* p) {
    return (unsigned)(uintptr_t)p;
}

// ---------------------------------------------------------------------------
// Single kernel:
//   1) async-DMA the 64x255 threshold tile into LDS (gfx1250 async-to-LDS)
//   2) build the 64x256 midpoint/dequant table in LDS from the staged tile
//   3) per element: branchless binary search (top 3 levels from registers,
//      bottom 5 levels from LDS) + midpoint gather, streamed with NT hints
// ---------------------------------------------------------------------------
__global__ __launch_bounds__(BLOCK) void qq_main_kernel(const float* __restrict__ x,
                                                        const float* __restrict__ thr,
                                                        float* __restrict__ out) {
    __shared__ __align__(16) float smem[FT * THR_STRIDE + FT * MID_STRIDE]; // 128 KB
    float* sthr = smem;                       // [FT][255] packed flat
    float* smid = smem + FT * THR_STRIDE;     // [FT][257] padded

    const int tid   = threadIdx.x;
    const int fbase = blockIdx.x * FT;
    const int rbase = blockIdx.y * ROWS_PER_BLOCK;

    // ---- async fill: thresholds tile, flat contiguous, 16B chunks ----
    {
        const float* gsrc = thr + (size_t)fbase * NT;   // 16320 floats, 16B aligned
        const int nvec = (FT * NT) / 4;                 // 4080
        for (int j = tid; j < nvec; j += BLOCK) {
            unsigned dst = lds_off(sthr + 4 * j);
            const float* src = gsrc + 4 * j;
            asm volatile("global_load_async_to_lds_b128 %0, %1, off"
                         :: "v"(dst), "v"(src) : "memory");
        }
    }
    asm volatile("s_wait_asynccnt 0" ::: "memory");  // my wave's DMAs done
    __syncthreads();                                  // everyone's DMAs done

    // ---- build midpoint table in LDS from the staged thresholds ----
    //   mid[0]   = 1.5*t0 - 0.5*t1
    //   mid[i]   = 0.5*(t[i-1] + t[i])   (1 <= i <= 254)
    //   mid[255] = 1.5*t254 - 0.5*t253
    for (int j = tid; j < FT * 256; j += BLOCK) {
        int fl = j >> 8, i = j & 255;
        const float* t = sthr + fl * THR_STRIDE;
        float m;
        if (i == 0)        m = 1.5f * t[0]      - 0.5f * t[1];
        else if (i == NT)  m = 1.5f * t[NT - 1] - 0.5f * t[NT - 2];
        else               m = 0.5f * (t[i - 1] + t[i]);
        smid[fl * MID_STRIDE + i] = m;
    }
    __syncthreads();

    const int fl   = tid & (FT - 1);                 // feature within tile
    const int rsub = tid >> 6;                       // 0..3 (BLOCK/FT rows per pass)
    const int RP   = BLOCK / FT;                     // 4
    const float* __restrict__ tf = sthr + fl * THR_STRIDE;
    const float* __restrict__ mf = smid + fl * MID_STRIDE;
    const int col = fbase + fl;

    // Top 3 levels of this thread's search tree, cached in registers:
    // step 128 probes t[127]; step 64 probes t[63]/t[191];
    // step 32 probes t[31]/t[95]/t[159]/t[223].
    const float p1  = tf[127];
    const float p2a = tf[63],  p2b = tf[191];
    const float p3a = tf[31],  p3b = tf[95], p3c = tf[159], p3d = tf[223];

    // 1024 rows / (4 rows-per-pass * unroll 4) = 64 iterations; 4 independent
    // search chains in flight per thread for ILP against ds_load latency.
    for (int r0 = rsub; r0 < ROWS_PER_BLOCK; r0 += RP * 4) {
        float v[4];
        int   lo[4];
#pragma unroll
        for (int u = 0; u < 4; ++u) {
            int idx = (rbase + r0 + u * RP) * NF + col;   // < 2^24, int is fine
            v[u]  = __builtin_nontemporal_load(x + idx);
            lo[u] = 0;
        }
        // Levels 1-3 from registers (no LDS traffic, short cndmask trees).
#pragma unroll
        for (int u = 0; u < 4; ++u) {
            int l = (p1 < v[u]) ? 128 : 0;
            float q2 = (l != 0) ? p2b : p2a;
            l += (q2 < v[u]) ? 64 : 0;
            float qlo = (l & 64) ? p3b : p3a;
            float qhi = (l & 64) ? p3d : p3c;
            float q3  = (l & 128) ? qhi : qlo;
            l += (q3 < v[u]) ? 32 : 0;
            lo[u] = l;
        }
        // Levels 4-8 from LDS: steps 16..1; candidate c = lo+s stays <= 255,
        // so tf[c-1] is always in range. Final lo == count(t < v)
        // == searchsorted(..., side='left').
#pragma unroll
        for (int s = 16; s >= 1; s >>= 1) {
#pragma unroll
            for (int u = 0; u < 4; ++u) {
                int c = lo[u] + s;
                lo[u] = (tf[c - 1] < v[u]) ? c : lo[u];
            }
        }
#pragma unroll
        for (int u = 0; u < 4; ++u) {
            int idx = (rbase + r0 + u * RP) * NF + col;
            __builtin_nontemporal_store(mf[lo[u]], out + idx);
        }
    }
}

extern "C" void kernel_launch(void* const* d_in, const int* in_sizes, int n_in,
                              void* d_out, int out_size, void* d_ws, size_t ws_size,
                              hipStream_t stream) {
    const float* x   = (const float*)d_in[0];   // [B, F] float32
    const float* thr = (const float*)d_in[1];   // [F, T] float32 (sorted per row)
    float* out = (float*)d_out;                 // [B, F] float32
    (void)d_ws; (void)ws_size; (void)in_sizes; (void)n_in; (void)out_size;

    // grid (F/FT, B/ROWS_PER_BLOCK) = (8, 32) = 256 blocks, 2 per WGP (128 KB LDS each).
    qq_main_kernel<<<dim3(NF / FT, NB / ROWS_PER_BLOCK), dim3(BLOCK), 0, stream>>>(
        x, thr, out);
}